// GivenPointSliceSamplingSplineWarpSSM_76218489635176
// MI455X (gfx1250) — compile-verified
//
#include <hip/hip_runtime.h>
#include <math.h>

// ---------------- constants ----------------
#define NCP     216
#define NPAD    224              // 216 padded to 14 tiles of 16
#define NSYS    220              // 216 + 4 (affine)
#define MCOLS   223              // 220 cols + 3 RHS
#define MSTRIDE 224
#define NSLICES 16
#define VDIM    128
#define PTOT    (NSLICES*VDIM*VDIM)   // 262144
#define VOLSZ   (VDIM*VDIM*VDIM)
#define EPSR    1e-10f

// ---------------- ws layout (floats) ----------------
#define WS_M    0
#define WS_WV   (WS_M + NSYS*MSTRIDE)     // 49280 : wv[224][4] (rows 216..219 = v)
#define WS_TP4  (WS_WV + NPAD*4)          // 50176 : t4[224][4] = (-2tx,-2ty,-2tz,|t|^2)
#define WS_R    (WS_TP4 + NPAD*4)         // 51072 : R[9]
#define WS_OFF  (WS_R + 16)               // 51088 : off[16][3] = global + slice shift
// total ~51136 floats (~200 KB)

typedef __attribute__((ext_vector_type(2))) float v2f;
typedef __attribute__((ext_vector_type(8))) float v8f;

__device__ __forceinline__ float fast_sqrtf(float x) {
  return __builtin_amdgcn_sqrtf(x);     // single v_sqrt_f32 (TRANS32, co-executes with VALU)
}

__device__ __forceinline__ void mat3mul(float* o, const float* a, const float* b) {
#pragma unroll
  for (int i = 0; i < 3; ++i)
#pragma unroll
    for (int j = 0; j < 3; ++j)
      o[i*3+j] = a[i*3+0]*b[0*3+j] + a[i*3+1]*b[1*3+j] + a[i*3+2]*b[2*3+j];
}

// ================= K0: rotation, offsets, t4 table, build LHS =================
__global__ __launch_bounds__(256) void k_setup(
    const float* __restrict__ wcp,   // (216,3) warped control points
    const float* __restrict__ gofs,  // (3)
    const float* __restrict__ pso,   // (16,3)
    const float* __restrict__ grot,  // (3)
    const float* __restrict__ cpts,  // (216,3)
    const float* __restrict__ csys,  // (16,3,3)
    const float* __restrict__ iar,   // (3)
    float* __restrict__ ws)
{
  const int tid = threadIdx.x;

  // rotation R = rz @ ry @ rx  (row-major, applied as q_d = sum_c g_c R[c][d])
  float th = grot[0] + iar[0], ph = grot[1] + iar[1], psn = grot[2] + iar[2];
  float ct = cosf(th),  st = sinf(th);
  float cp = cosf(ph),  sp = sinf(ph);
  float cs = cosf(psn), ss = sinf(psn);
  float rx[9] = {1,0,0,  0,ct,st,  0,-st,ct};
  float ry[9] = {cp,0,-sp,  0,1,0,  sp,0,cp};
  float rz[9] = {cs,ss,0,  -ss,cs,0,  0,0,1};
  float rzy[9], R[9];
  mat3mul(rzy, rz, ry);
  mat3mul(R, rzy, rx);

  if (tid == 0) {
#pragma unroll
    for (int i = 0; i < 9; ++i) ws[WS_R + i] = R[i];
  }

  // off[s][k] = global_offset[k] + sum_i pso[s][i] * (csys[s] @ R)[i][k]
  if (tid < NSLICES) {
    int s = tid;
#pragma unroll
    for (int k = 0; k < 3; ++k) {
      float sh = 0.f;
#pragma unroll
      for (int i = 0; i < 3; ++i) {
        float rc = csys[s*9+i*3+0]*R[0*3+k] + csys[s*9+i*3+1]*R[1*3+k] + csys[s*9+i*3+2]*R[2*3+k];
        sh = fmaf(pso[s*3+i], rc, sh);
      }
      ws[WS_OFF + s*3 + k] = gofs[k] + sh;
    }
  }

  // t4 table (B matrix columns), rows >= 216 zero-padded
  if (tid < NPAD) {
    float t0 = 0.f, t1 = 0.f, t2 = 0.f, sq = 0.f;
    if (tid < NCP) {
      t0 = 2.f*wcp[tid*3+2] - 1.f;   // tp = flip(wcp)*2-1
      t1 = 2.f*wcp[tid*3+1] - 1.f;
      t2 = 2.f*wcp[tid*3+0] - 1.f;
      sq = t0*t0 + t1*t1 + t2*t2;
    }
    ws[WS_TP4 + tid*4 + 0] = -2.f*t0;
    ws[WS_TP4 + tid*4 + 1] = -2.f*t1;
    ws[WS_TP4 + tid*4 + 2] = -2.f*t2;
    ws[WS_TP4 + tid*4 + 3] = sq;
  }

  // Build augmented LHS: [[A, B],[B^T, 0]] | [tv; 0]
  for (int idx = tid; idx < NSYS*MCOLS; idx += 256) {
    int i = idx / MCOLS, j = idx % MCOLS;
    float val;
    if (i < NCP) {
      float tix = 2.f*wcp[i*3+2]-1.f, tiy = 2.f*wcp[i*3+1]-1.f, tiz = 2.f*wcp[i*3+0]-1.f;
      if (j < NCP) {
        float tjx = 2.f*wcp[j*3+2]-1.f, tjy = 2.f*wcp[j*3+1]-1.f, tjz = 2.f*wcp[j*3+0]-1.f;
        float dx = tix-tjx, dy = tiy-tjy, dz = tiz-tjz;
        val = fast_sqrtf(fmaxf(dx*dx + dy*dy + dz*dz, EPSR));
      } else if (j == 216) val = tix;
      else if (j == 217) val = tiy;
      else if (j == 218) val = tiz;
      else if (j == 219) val = 1.f;
      else {             // RHS: tv = flip(cpts)*2-1
        int k = j - 220;
        val = 2.f*cpts[i*3 + (2 - k)] - 1.f;
      }
    } else {
      int d = i - 216;
      if (j < NCP) val = (d < 3) ? (2.f*wcp[j*3 + (2 - d)] - 1.f) : 1.f;
      else         val = 0.f;
    }
    ws[WS_M + i*MSTRIDE + j] = val;
  }
}

// ================= K1: Gauss-Jordan solve with partial pivoting =================
__global__ __launch_bounds__(256) void k_solve(float* __restrict__ ws)
{
  __shared__ float rmax[256];
  __shared__ int   ridx[256];
  const int tid = threadIdx.x;
  float* M = ws + WS_M;

  for (int k = 0; k < NSYS; ++k) {
    // --- partial pivot search over column k, rows k..219 ---
    float best = -1.f; int bi = k;
    for (int i = k + tid; i < NSYS; i += 256) {
      float v = fabsf(M[i*MSTRIDE + k]);
      if (v > best) { best = v; bi = i; }
    }
    rmax[tid] = best; ridx[tid] = bi;
    __syncthreads();
    for (int s = 128; s > 0; s >>= 1) {
      if (tid < s && rmax[tid + s] > rmax[tid]) { rmax[tid] = rmax[tid + s]; ridx[tid] = ridx[tid + s]; }
      __syncthreads();
    }
    int pr = ridx[0];
    // --- swap rows k <-> pr (cols k..222) ---
    if (pr != k) {
      for (int j = k + tid; j < MCOLS; j += 256) {
        float a = M[k*MSTRIDE + j], b = M[pr*MSTRIDE + j];
        M[k*MSTRIDE + j] = b; M[pr*MSTRIDE + j] = a;
      }
    }
    __syncthreads();
    float piv = M[k*MSTRIDE + k];
    // --- eliminate column k from all other rows ---
    for (int i = tid; i < NSYS; i += 256) {
      if (i != k) {
        float f = M[i*MSTRIDE + k] / piv;
        for (int j = k; j < MCOLS; ++j)
          M[i*MSTRIDE + j] = fmaf(-f, M[k*MSTRIDE + j], M[i*MSTRIDE + j]);
      }
    }
    __syncthreads();
  }

  // extract solution: x[i][c] = rhs'[i][c] / diag[i]; rows >= 220 padded to 0
  if (tid < NPAD) {
    float d = (tid < NSYS) ? M[tid*MSTRIDE + tid] : 1.f;
#pragma unroll
    for (int k = 0; k < 3; ++k)
      ws[WS_WV + tid*4 + k] = (tid < NSYS) ? (M[tid*MSTRIDE + 220 + k] / d) : 0.f;
    ws[WS_WV + tid*4 + 3] = 0.f;
  }
}

// ================= K2: fused TPS evaluation (WMMA) + trilinear sample =================
__global__ __launch_bounds__(256) void k_warp_sample(
    const float* __restrict__ grid,
    const float* __restrict__ vol,
    const float* __restrict__ ws,
    float* __restrict__ out)
{
  __shared__ float s_tp4[NPAD*4];
  __shared__ float s_wv[NPAD*4];
  __shared__ float s_R[9];
  __shared__ float s_off[NSLICES*3];
  __shared__ float s_loc[8][16][3];

  const int tid = threadIdx.x;
  for (int i = tid; i < NPAD*4; i += 256) { s_tp4[i] = ws[WS_TP4 + i]; s_wv[i] = ws[WS_WV + i]; }
  if (tid < 9)          s_R[tid]   = ws[WS_R + tid];
  if (tid < NSLICES*3)  s_off[tid] = ws[WS_OFF + tid];
  __syncthreads();

  const int wave = tid >> 5, lane = tid & 31;
  const int h = lane >> 4, lm = lane & 15;
  const int tileBase = (blockIdx.x * 8 + wave) * 16;
  const int p = tileBase + lm;               // both lane halves hold the same point lm
  const int sidx = p >> 14;                  // slice (16 points never cross a slice)

  // rotated + shifted query point
  float gx = grid[3*p+0], gy = grid[3*p+1], gz = grid[3*p+2];
  float qx = gx*s_R[0] + gy*s_R[3] + gz*s_R[6] + s_off[sidx*3+0];
  float qy = gx*s_R[1] + gy*s_R[4] + gz*s_R[7] + s_off[sidx*3+1];
  float qz = gx*s_R[2] + gy*s_R[5] + gz*s_R[8] + s_off[sidx*3+2];
  float qsq = qx*qx + qy*qy + qz*qz;

  // A fragment (16x4 f32): lane holds row lm, K = {2h, 2h+1} of [qx,qy,qz,1]
  v2f a;
  a.x = h ? qz : qx;
  a.y = h ? 1.0f : qy;

  // C fragment: c[r] = |q|^2 of row (r + 8h)  (broadcast via shfl)
  v8f cfrag;
#pragma unroll
  for (int r = 0; r < 8; ++r) cfrag[r] = __shfl(qsq, r + 8*h, 32);

  float accx[8], accy[8], accz[8];
#pragma unroll
  for (int r = 0; r < 8; ++r) { accx[r] = 0.f; accy[r] = 0.f; accz[r] = 0.f; }

  // 14 tiles over 224 (zero-padded) control points:
  // D = A*B + C = |q|^2 - 2 q.t + |t|^2 = squared distance tile (full fp32)
#pragma unroll
  for (int nt = 0; nt < NPAD/16; ++nt) {
    int n = nt*16 + lm;
    v2f b;
    b.x = s_tp4[n*4 + 2*h + 0];
    b.y = s_tp4[n*4 + 2*h + 1];
    v8f d = __builtin_amdgcn_wmma_f32_16x16x4_f32(
        false, a, false, b, (short)0, cfrag, false, false);
    float w0 = s_wv[n*4+0], w1 = s_wv[n*4+1], w2 = s_wv[n*4+2];
#pragma unroll
    for (int r = 0; r < 8; ++r) {
      float kq = fast_sqrtf(fmaxf(d[r], EPSR));   // phi: single v_sqrt_f32
      accx[r] = fmaf(kq, w0, accx[r]);
      accy[r] = fmaf(kq, w1, accy[r]);
      accz[r] = fmaf(kq, w2, accz[r]);
    }
  }

  // reduce the N dimension across each 16-lane half (xor 1,2,4,8 stays in half)
#pragma unroll
  for (int r = 0; r < 8; ++r) {
#pragma unroll
    for (int s = 1; s <= 8; s <<= 1) {
      accx[r] += __shfl_xor(accx[r], s, 32);
      accy[r] += __shfl_xor(accy[r], s, 32);
      accz[r] += __shfl_xor(accz[r], s, 32);
    }
  }

  // add affine part Q@v  (v rows live at s_wv[216..219])
  float lrx[8], lry[8], lrz[8];
#pragma unroll
  for (int r = 0; r < 8; ++r) {
    int m = r + 8*h;
    float rqx = __shfl(qx, m, 32);
    float rqy = __shfl(qy, m, 32);
    float rqz = __shfl(qz, m, 32);
    lrx[r] = accx[r] + rqx*s_wv[216*4+0] + rqy*s_wv[217*4+0] + rqz*s_wv[218*4+0] + s_wv[219*4+0];
    lry[r] = accy[r] + rqx*s_wv[216*4+1] + rqy*s_wv[217*4+1] + rqz*s_wv[218*4+1] + s_wv[219*4+1];
    lrz[r] = accz[r] + rqx*s_wv[216*4+2] + rqy*s_wv[217*4+2] + rqz*s_wv[218*4+2] + s_wv[219*4+2];
  }
  if (lm == 0) {           // lane 0 -> rows 0..7, lane 16 -> rows 8..15
#pragma unroll
    for (int r = 0; r < 8; ++r) {
      int m = r + 8*h;
      s_loc[wave][m][0] = lrx[r];
      s_loc[wave][m][1] = lry[r];
      s_loc[wave][m][2] = lrz[r];
    }
  }
  __syncthreads();

  // ---------- trilinear sample: lane handles (point lm, channels 2h, 2h+1) ----------
  float lx = s_loc[wave][lm][0], ly = s_loc[wave][lm][1], lz = s_loc[wave][lm][2];
  float x = (lx + 1.f) * 0.5f * (float)(VDIM - 1);
  float y = (ly + 1.f) * 0.5f * (float)(VDIM - 1);
  float z = (lz + 1.f) * 0.5f * (float)(VDIM - 1);
  float x0f = floorf(x), y0f = floorf(y), z0f = floorf(z);
  int x0 = (int)x0f, y0 = (int)y0f, z0 = (int)z0f;
  float xd = x - x0f, yd = y - y0f, zd = z - z0f;

  const int c0 = 2*h;
  const float* v0 = vol + (size_t)c0 * VOLSZ;
  const float* v1 = v0 + VOLSZ;
  float a0 = 0.f, a1 = 0.f;
#pragma unroll
  for (int corner = 0; corner < 8; ++corner) {
    int dz = corner >> 2, dy = (corner >> 1) & 1, dx = corner & 1;
    int zi = z0 + dz, yi = y0 + dy, xi = x0 + dx;
    float wz = dz ? zd : 1.f - zd;
    float wy = dy ? yd : 1.f - yd;
    float wx = dx ? xd : 1.f - xd;
    bool valid = (zi >= 0) & (zi < VDIM) & (yi >= 0) & (yi < VDIM) & (xi >= 0) & (xi < VDIM);
    int zc = min(max(zi, 0), VDIM-1);
    int yc = min(max(yi, 0), VDIM-1);
    int xc = min(max(xi, 0), VDIM-1);
    int idx = (zc*VDIM + yc)*VDIM + xc;
    float w = wz*wy*wx * (valid ? 1.f : 0.f);
    a0 = fmaf(w, v0[idx], a0);
    a1 = fmaf(w, v1[idx], a1);
  }

  // output (1, C, 128, 128, 16): out[c, yy, xx, s]
  int rem = p & 16383;
  int yy = rem >> 7, xx = rem & 127;
  size_t obase = ((size_t)yy*VDIM + xx)*NSLICES + sidx;
  size_t cstride = (size_t)VDIM*VDIM*NSLICES;
  out[(size_t)c0*cstride + obase]     = a0;
  out[(size_t)(c0+1)*cstride + obase] = a1;
}

// ================= host launcher =================
extern "C" void kernel_launch(void* const* d_in, const int* in_sizes, int n_in,
                              void* d_out, int out_size, void* d_ws, size_t ws_size,
                              hipStream_t stream) {
  const float* wcp  = (const float*)d_in[0];  // (1,216,3)
  const float* vol  = (const float*)d_in[1];  // (1,4,128,128,128)
  const float* gofs = (const float*)d_in[2];  // (1,1,3)
  const float* pso  = (const float*)d_in[3];  // (1,16,3)
  const float* grot = (const float*)d_in[4];  // (1,3)
  const float* cpts = (const float*)d_in[5];  // (216,3)
  const float* grid = (const float*)d_in[6];  // (1,P,3)
  const float* csys = (const float*)d_in[7];  // (16,3,3)
  const float* iar  = (const float*)d_in[8];  // (3,)
  float* out = (float*)d_out;
  float* ws  = (float*)d_ws;

  (void)in_sizes; (void)n_in; (void)out_size; (void)ws_size;

  k_setup<<<1, 256, 0, stream>>>(wcp, gofs, pso, grot, cpts, csys, iar, ws);
  k_solve<<<1, 256, 0, stream>>>(ws);
  k_warp_sample<<<PTOT/(16*8), 256, 0, stream>>>(grid, vol, ws, out);
}